// GCN_13039520711474
// MI455X (gfx1250) — compile-verified
//
#include <hip/hip_runtime.h>

typedef __attribute__((ext_vector_type(2))) float v2f;
typedef __attribute__((ext_vector_type(8))) float v8f;

#define N_NODES  50000
#define N_EDGES  800000
#define N_GRAPHS 32
#define IN_C     128
#define HID_C    128
#define OUT_C    64

// ================= CSR build (once, reused by both layers) =================
__global__ void k_izero(int* __restrict__ p, int n) {
    int i = blockIdx.x * blockDim.x + threadIdx.x;
    if (i < n) p[i] = 0;
}

__global__ void k_ideg_count(const int* __restrict__ dst, int* __restrict__ ideg, int nE) {
    int i = blockIdx.x * blockDim.x + threadIdx.x;
    if (i < nE) atomicAdd(&ideg[dst[i]], 1);
}

__global__ void k_dinv(const int* __restrict__ ideg, float* __restrict__ dinv, int n) {
    int i = blockIdx.x * blockDim.x + threadIdx.x;
    if (i < n) dinv[i] = rsqrtf((float)ideg[i] + 1.0f);   // +1 self-loop
}

// Single-workgroup exclusive scan: row_start[0..n] from ideg[0..n-1].
// 1024 threads, each owns a contiguous chunk; Hillis-Steele over per-thread sums in LDS.
__global__ void k_scan(const int* __restrict__ ideg, int* __restrict__ row_start, int n) {
    __shared__ int sums[1024];
    const int t = threadIdx.x;
    const int chunk = (n + 1023) / 1024;
    const int begin = t * chunk;
    const int end   = min(begin + chunk, n);
    int s = 0;
    for (int i = begin; i < end; ++i) s += ideg[i];
    sums[t] = s;
    __syncthreads();
    for (int off = 1; off < 1024; off <<= 1) {
        int v = (t >= off) ? sums[t - off] : 0;
        __syncthreads();
        sums[t] += v;
        __syncthreads();
    }
    int prefix = (t == 0) ? 0 : sums[t - 1];
    for (int i = begin; i < end; ++i) {
        int dv = ideg[i];
        row_start[i] = prefix;
        prefix += dv;
    }
    if (t == 0) row_start[n] = sums[1023];
}

__global__ void k_icopy(const int* __restrict__ a, int* __restrict__ b, int n) {
    int i = blockIdx.x * blockDim.x + threadIdx.x;
    if (i < n) b[i] = a[i];
}

__global__ void k_csr_fill(const int* __restrict__ src, const int* __restrict__ dst,
                           int* __restrict__ cursor, int* __restrict__ csr_src, int nE) {
    int e = blockIdx.x * blockDim.x + threadIdx.x;
    if (e < nE) {
        int pos = atomicAdd(&cursor[dst[e]], 1);
        csr_src[pos] = src[e];
    }
}

// ============ fp32 WMMA GEMM: C[M,128] = A[M,128] * B[128,128] ============
// One wave = one 16x16 tile via 32 chained V_WMMA_F32_16X16X4_F32.
// 8 waves/block cover the 128 output channels of a 16-row stripe. EXEC all-ones.
__global__ void k_gemm128_wmma(const float* __restrict__ A, const float* __restrict__ B,
                               float* __restrict__ C) {
    const int lane = threadIdx.x & 31;
    const int wave = threadIdx.x >> 5;
    const int row0 = blockIdx.x * 16;
    const int col0 = wave * 16;
    const int m    = lane & 15;
    const int half = lane >> 4;

    v8f acc = {};
    const float* arow = A + (size_t)(row0 + m) * 128;
    const float* bcol = B + col0 + m;

    #pragma unroll 4
    for (int s = 0; s < 32; ++s) {
        const int k = s * 4 + half * 2;
        v2f a, b;
        a.x = arow[k];
        a.y = arow[k + 1];
        b.x = bcol[(size_t)k * 128];
        b.y = bcol[(size_t)(k + 1) * 128];
        acc = __builtin_amdgcn_wmma_f32_16x16x4_f32(
                  false, a, false, b, (short)0, acc, false, false);
    }

    float* crow = C + col0 + m;
    #pragma unroll
    for (int v = 0; v < 8; ++v)
        crow[(size_t)(row0 + v + half * 8) * 128] = acc[v];
}

// ==== fused aggregation (gather form): out[i] = relu(b + sum_j norm*xw[j]) ====
// One wave per destination node; 4 channels/lane in registers; NO float atomics.
// Self-loop folded in as xw[i]*dinv[i]^2. xw (25.6MB) is L2-resident on MI455X (192MB L2).
__global__ void k_gather_agg(const float* __restrict__ xw, const int* __restrict__ row_start,
                             const int* __restrict__ csr_src, const float* __restrict__ dinv,
                             const float* __restrict__ bias, float* __restrict__ out, int n) {
    const int lane = threadIdx.x & 31;
    const int i    = (blockIdx.x * blockDim.x + threadIdx.x) >> 5;
    if (i >= n) return;

    const float di = dinv[i];
    float4 acc = ((const float4*)(xw + (size_t)i * 128))[lane];     // self-loop row
    const float w_self = di * di;
    acc.x *= w_self; acc.y *= w_self; acc.z *= w_self; acc.w *= w_self;

    const int e1 = row_start[i + 1];
    int e = row_start[i];
    for (; e + 1 < e1; e += 2) {                                    // 2-way to hide gather latency
        const int s0 = csr_src[e];
        const int s1 = csr_src[e + 1];
        const float w0 = dinv[s0] * di;
        const float w1 = dinv[s1] * di;
        const float4 v0 = ((const float4*)(xw + (size_t)s0 * 128))[lane];
        const float4 v1 = ((const float4*)(xw + (size_t)s1 * 128))[lane];
        acc.x += v0.x * w0 + v1.x * w1;
        acc.y += v0.y * w0 + v1.y * w1;
        acc.z += v0.z * w0 + v1.z * w1;
        acc.w += v0.w * w0 + v1.w * w1;
    }
    if (e < e1) {
        const int s0 = csr_src[e];
        const float w0 = dinv[s0] * di;
        const float4 v0 = ((const float4*)(xw + (size_t)s0 * 128))[lane];
        acc.x += v0.x * w0; acc.y += v0.y * w0; acc.z += v0.z * w0; acc.w += v0.w * w0;
    }

    const float4 bv = ((const float4*)bias)[lane];
    float4 r;
    r.x = fmaxf(acc.x + bv.x, 0.0f);
    r.y = fmaxf(acc.y + bv.y, 0.0f);
    r.z = fmaxf(acc.z + bv.z, 0.0f);
    r.w = fmaxf(acc.w + bv.w, 0.0f);
    ((float4*)(out + (size_t)i * 128))[lane] = r;
}

// ================= pooling (batch is sorted: chunked accumulate) =================
__global__ void k_zero(float* __restrict__ p, int n) {
    int i = blockIdx.x * blockDim.x + threadIdx.x;
    if (i < n) p[i] = 0.0f;
}

__device__ __forceinline__ void pool_flush(float* pooled, float* cnt, int g, int lane,
                                           const float4& acc, float c) {
    float* q = pooled + g * 128 + lane * 4;
    atomicAdd(q + 0, acc.x);
    atomicAdd(q + 1, acc.y);
    atomicAdd(q + 2, acc.z);
    atomicAdd(q + 3, acc.w);
    if (lane == 0) atomicAdd(&cnt[g], c);
}

__global__ void k_pool(const float* __restrict__ h, const int* __restrict__ batch,
                       float* __restrict__ pooled, float* __restrict__ cnt, int n, int chunk) {
    const int lane = threadIdx.x & 31;
    const int w    = (blockIdx.x * blockDim.x + threadIdx.x) >> 5;
    const int begin = w * chunk;
    if (begin >= n) return;
    const int end = min(begin + chunk, n);

    int g = batch[begin];
    float4 acc = make_float4(0.f, 0.f, 0.f, 0.f);
    float c = 0.f;
    for (int i = begin; i < end; ++i) {
        const int gi = batch[i];                   // wave-uniform
        if (gi != g) {
            pool_flush(pooled, cnt, g, lane, acc, c);
            acc = make_float4(0.f, 0.f, 0.f, 0.f);
            c = 0.f;
            g = gi;
        }
        const float4 v = ((const float4*)(h + (size_t)i * 128))[lane];
        acc.x += v.x; acc.y += v.y; acc.z += v.z; acc.w += v.w;
        c += 1.f;
    }
    pool_flush(pooled, cnt, g, lane, acc, c);
}

// ================= final FC =================
__global__ void k_fc(const float* __restrict__ pooled, const float* __restrict__ cnt,
                     const float* __restrict__ Wfc, const float* __restrict__ bfc,
                     float* __restrict__ out) {
    int idx = blockIdx.x * blockDim.x + threadIdx.x;
    if (idx < N_GRAPHS * OUT_C) {
        const int g = idx / OUT_C;
        const int o = idx % OUT_C;
        const float inv = 1.0f / fmaxf(cnt[g], 1.0f);
        const float* pr = pooled + g * 128;
        float acc = bfc[o];
        #pragma unroll 8
        for (int c = 0; c < 128; ++c)
            acc = fmaf(pr[c] * inv, Wfc[c * OUT_C + o], acc);
        out[idx] = acc;
    }
}

extern "C" void kernel_launch(void* const* d_in, const int* in_sizes, int n_in,
                              void* d_out, int out_size, void* d_ws, size_t ws_size,
                              hipStream_t stream) {
    const float* x     = (const float*)d_in[0];
    const int*   ei    = (const int*)  d_in[1];   // [2, N_EDGES]
    const int*   batch = (const int*)  d_in[2];
    const float* W1    = (const float*)d_in[3];
    const float* b1    = (const float*)d_in[4];
    const float* W2    = (const float*)d_in[5];
    const float* b2    = (const float*)d_in[6];
    const float* Wfc   = (const float*)d_in[7];
    const float* bfc   = (const float*)d_in[8];
    float* out = (float*)d_out;

    const int* src = ei;
    const int* dst = ei + N_EDGES;

    // ---- workspace layout ----
    float* ws        = (float*)d_ws;
    float* dinv      = ws;                                   // 50000 f
    float* bufA      = dinv + 50000;                         // N*128 f
    float* bufB      = bufA + (size_t)N_NODES * 128;         // N*128 f
    float* pooled    = bufB + (size_t)N_NODES * 128;         // 32*128 f
    float* cnt       = pooled + N_GRAPHS * 128;              // 32 f
    int*   ideg      = (int*)(cnt + 32);                     // N
    int*   row_start = ideg + N_NODES;                       // N+1
    int*   cursor    = row_start + N_NODES + 1;              // N
    int*   csr_src   = cursor + N_NODES;                     // E

    const int BLK = 256;
    const dim3 gN((N_NODES + BLK - 1) / BLK);
    const dim3 gE((N_EDGES + BLK - 1) / BLK);
    const dim3 gGemm(N_NODES / 16);                          // 3125 blocks, 8 waves each
    const dim3 gAgg(N_NODES / 8);                            // 6250 blocks, 1 wave/node

    // ---- CSR build + normalization (once) ----
    k_izero     <<<gN, BLK, 0, stream>>>(ideg, N_NODES);
    k_ideg_count<<<gE, BLK, 0, stream>>>(dst, ideg, N_EDGES);
    k_dinv      <<<gN, BLK, 0, stream>>>(ideg, dinv, N_NODES);
    k_scan      <<<1, 1024, 0, stream>>>(ideg, row_start, N_NODES);
    k_icopy     <<<gN, BLK, 0, stream>>>(row_start, cursor, N_NODES);
    k_csr_fill  <<<gE, BLK, 0, stream>>>(src, dst, cursor, csr_src, N_EDGES);

    // ---- layer 1: h1 = relu(Agg(x @ W1) + b1) ----
    k_gemm128_wmma<<<gGemm, BLK, 0, stream>>>(x, W1, bufA);
    k_gather_agg  <<<gAgg, BLK, 0, stream>>>(bufA, row_start, csr_src, dinv, b1, bufB, N_NODES);

    // ---- layer 2: h2 = relu(Agg(h1 @ W2) + b2) ----
    k_gemm128_wmma<<<gGemm, BLK, 0, stream>>>(bufB, W2, bufA);
    k_gather_agg  <<<gAgg, BLK, 0, stream>>>(bufA, row_start, csr_src, dinv, b2, bufB, N_NODES);

    // ---- global mean pool + FC ----
    k_zero<<<(N_GRAPHS * 128 + N_GRAPHS + BLK - 1) / BLK, BLK, 0, stream>>>(pooled, N_GRAPHS * 128 + N_GRAPHS);
    {
        const int waves = 2048;                              // 256 blocks * 8 waves
        const int chunk = (N_NODES + waves - 1) / waves;     // ~25 sorted nodes/wave
        k_pool<<<256, BLK, 0, stream>>>(bufB, batch, pooled, cnt, N_NODES, chunk);
    }
    k_fc<<<(N_GRAPHS * OUT_C + BLK - 1) / BLK, BLK, 0, stream>>>(pooled, cnt, Wfc, bfc, out);
}